// HybridSamplerQuanvolution_65481071404120
// MI455X (gfx1250) — compile-verified
//
#include <hip/hip_runtime.h>
#include <hip/hip_bf16.h>
#include <math.h>

typedef __attribute__((ext_vector_type(2))) float v2f;
typedef __attribute__((ext_vector_type(8))) float v8f;

// ---------------------------------------------------------------------------
// Kernel 1: 2-qubit sampler circuit (all RY + CNOT => purely real state).
// probs[:,0] -> a0, probs[:,1] -> a1, stored as (B,2).
// ---------------------------------------------------------------------------
__global__ void sampler_kernel(const float* __restrict__ inp,   // (B,2)
                               const float* __restrict__ w,     // (2,)
                               float* __restrict__ a01, int B) {
  int b = blockIdx.x * blockDim.x + threadIdx.x;
  if (b >= B) return;
  float st0 = 1.f, st1 = 0.f, st2 = 0.f, st3 = 0.f;
  float c, sn, a, bb, tm;

  // RY(inp0) on wire 1 (stride 1): pairs (0,1),(2,3)
  sincosf(0.5f * inp[2 * b + 0], &sn, &c);
  a = st0; bb = st1; st0 = c * a - sn * bb; st1 = sn * a + c * bb;
  a = st2; bb = st3; st2 = c * a - sn * bb; st3 = sn * a + c * bb;
  // RY(inp1) on wire 0 (stride 2): pairs (0,2),(1,3)
  sincosf(0.5f * inp[2 * b + 1], &sn, &c);
  a = st0; bb = st2; st0 = c * a - sn * bb; st2 = sn * a + c * bb;
  a = st1; bb = st3; st1 = c * a - sn * bb; st3 = sn * a + c * bb;
  // CNOT(ctrl=wire1 (bit0), tgt=wire0 (bit1)): swap st1 <-> st3
  tm = st1; st1 = st3; st3 = tm;
  // RY(w0) wire 1
  sincosf(0.5f * w[0], &sn, &c);
  a = st0; bb = st1; st0 = c * a - sn * bb; st1 = sn * a + c * bb;
  a = st2; bb = st3; st2 = c * a - sn * bb; st3 = sn * a + c * bb;
  // RY(w1) wire 0
  sincosf(0.5f * w[1], &sn, &c);
  a = st0; bb = st2; st0 = c * a - sn * bb; st2 = sn * a + c * bb;
  a = st1; bb = st3; st1 = c * a - sn * bb; st3 = sn * a + c * bb;
  // CNOT(1,0)
  tm = st1; st1 = st3; st3 = tm;

  a01[2 * b + 0] = st0 * st0;   // probs[:,0]
  a01[2 * b + 1] = st1 * st1;   // probs[:,1]
}

// ---------------------------------------------------------------------------
// 4-qubit gate helpers. Wire w (0..3, wire 0 = MSB) -> stride S = 8 >> w.
// Fully unrolled, compile-time strides: state stays in VGPRs.
// ---------------------------------------------------------------------------
template <int S>
__device__ __forceinline__ void g_ry(float* re, float* im, float c, float s) {
#pragma unroll
  for (int i = 0; i < 16; ++i) {
    if (i & S) continue;
    const int j = i | S;
    float r0 = re[i], i0 = im[i], r1 = re[j], i1 = im[j];
    re[i] = c * r0 - s * r1;  im[i] = c * i0 - s * i1;
    re[j] = s * r0 + c * r1;  im[j] = s * i0 + c * i1;
  }
}
template <int S>
__device__ __forceinline__ void g_rx(float* re, float* im, float c, float s) {
#pragma unroll
  for (int i = 0; i < 16; ++i) {
    if (i & S) continue;
    const int j = i | S;
    float r0 = re[i], i0 = im[i], r1 = re[j], i1 = im[j];
    re[i] = c * r0 + s * i1;   im[i] = c * i0 - s * r1;
    re[j] = s * i0 + c * r1;   im[j] = -s * r0 + c * i1;
  }
}
template <int S>
__device__ __forceinline__ void g_rz(float* re, float* im, float c, float s) {
#pragma unroll
  for (int i = 0; i < 16; ++i) {
    if (i & S) continue;
    const int j = i | S;
    float r0 = re[i], i0 = im[i], r1 = re[j], i1 = im[j];
    re[i] = c * r0 + s * i0;  im[i] = c * i0 - s * r0;   // * e^{-i t/2}
    re[j] = c * r1 - s * i1;  im[j] = c * i1 + s * r1;   // * e^{+i t/2}
  }
}
template <int CB, int TB>
__device__ __forceinline__ void g_cnot(float* re, float* im) {
#pragma unroll
  for (int i = 0; i < 16; ++i) {
    if ((i & CB) && !(i & TB)) {
      const int j = i | TB;
      float t = re[i]; re[i] = re[j]; re[j] = t;
      t = im[i]; im[i] = im[j]; im[j] = t;
    }
  }
}

// ---------------------------------------------------------------------------
// Kernel 2: one thread per (batch, patch). Builds the real product state from
// the fused RY prefix (RY(px)+RY(a) on the same wire add angles), applies the
// fixed 8-op layer, emits 4 <Z_k> features.
// ---------------------------------------------------------------------------
__global__ void patch_kernel(const float* __restrict__ x,     // (B,784)
                             const float* __restrict__ a01,   // (B,2)
                             const float* __restrict__ qp,    // (6,)
                             float* __restrict__ F, int B) {
  int t = blockIdx.x * blockDim.x + threadIdx.x;
  if (t >= B * 196) return;
  int b = t / 196, p = t % 196;
  int pr = p / 14, pc = p % 14;
  const float* xb = x + (size_t)b * 784;
  float th0 = xb[(2 * pr) * 28 + 2 * pc]     + a01[2 * b + 0];
  float th1 = xb[(2 * pr) * 28 + 2 * pc + 1] + a01[2 * b + 1];
  float th2 = xb[(2 * pr + 1) * 28 + 2 * pc];
  float th3 = xb[(2 * pr + 1) * 28 + 2 * pc + 1];

  float f0[2], f1[2], f2[2], f3[2];
  sincosf(0.5f * th0, &f0[1], &f0[0]);
  sincosf(0.5f * th1, &f1[1], &f1[0]);
  sincosf(0.5f * th2, &f2[1], &f2[0]);
  sincosf(0.5f * th3, &f3[1], &f3[0]);

  float re[16], im[16];
#pragma unroll
  for (int i = 0; i < 16; ++i) {
    re[i] = f0[(i >> 3) & 1] * f1[(i >> 2) & 1] * f2[(i >> 1) & 1] * f3[i & 1];
    im[i] = 0.f;
  }

  float c, s;
  sincosf(0.5f * qp[0], &s, &c); g_rx<8>(re, im, c, s);   // RX(p0) wire 0
  sincosf(0.5f * qp[1], &s, &c); g_ry<4>(re, im, c, s);   // RY(p1) wire 1
  sincosf(0.5f * qp[2], &s, &c); g_rz<2>(re, im, c, s);   // RZ(p2) wire 2
  sincosf(0.5f * qp[3], &s, &c); g_rx<1>(re, im, c, s);   // RX(p3) wire 3
  g_cnot<8, 2>(re, im);                                   // CNOT(0,2)
  sincosf(0.5f * qp[4], &s, &c); g_ry<4>(re, im, c, s);   // RY(p4) wire 1
  sincosf(0.5f * qp[5], &s, &c); g_rz<1>(re, im, c, s);   // RZ(p5) wire 3
  g_cnot<4, 1>(re, im);                                   // CNOT(1,3)

  float z0 = 0.f, z1 = 0.f, z2 = 0.f, z3 = 0.f;
#pragma unroll
  for (int i = 0; i < 16; ++i) {
    float pi = re[i] * re[i] + im[i] * im[i];
    z0 += (i & 8) ? -pi : pi;
    z1 += (i & 4) ? -pi : pi;
    z2 += (i & 2) ? -pi : pi;
    z3 += (i & 1) ? -pi : pi;
  }
  float* out = F + (size_t)b * 784 + (size_t)p * 4;
  out[0] = z0; out[1] = z1; out[2] = z2; out[3] = z3;
}

// ---------------------------------------------------------------------------
// Kernel 3: logits = F(Bx784) @ W^T(784x10) + b via V_WMMA_F32_16X16X4_F32.
// 8 waves / block; W is staged once into LDS per block, zero-padded to 16
// columns at fill time (no mask ops in the 196-step K loop). LDS row stride
// 788 floats: 788 mod 64 = 20, gcd(20,64)=4 -> 16 consecutive lanes hit 16
// distinct banks for the ds_load_b64 B-fragment fetch.
// A layout: lane<16 -> (M=lane, K={0,1}); lane>=16 -> (M=lane-16, K={2,3}).
// C/D layout: element (m,n) in vgpr m%8, lane (m/8)*16 + n.
// ---------------------------------------------------------------------------
#define W_LDS_STRIDE 788

__global__ void gemm_wmma_kernel(const float* __restrict__ F,
                                 const float* __restrict__ W,
                                 const float* __restrict__ bias,
                                 float* __restrict__ L, int B) {
  __shared__ float ldsW[16 * W_LDS_STRIDE];   // 50,432 B of 320 KB WGP LDS

  int tid = threadIdx.x;
  // Cooperative fill: rows 0..9 = W, rows 10..15 = 0 (static N-padding).
  for (int idx = tid; idx < 16 * 784; idx += blockDim.x) {
    int n = idx / 784, k = idx - n * 784;
    ldsW[n * W_LDS_STRIDE + k] = (n < 10) ? W[n * 784 + k] : 0.0f;
  }
  __syncthreads();

  int wave = tid >> 5;
  int lane = tid & 31;
  int half = lane >> 4;        // 0: K {0,1}, 1: K {2,3}
  int ln   = lane & 15;        // M (for A) / N (for B,C)
  int row0 = (blockIdx.x * (blockDim.x >> 5) + wave) * 16;

  v8f c;
  float cval = (ln < 10) ? bias[ln] : 0.0f;
#pragma unroll
  for (int r = 0; r < 8; ++r) c[r] = cval;

  int arow_idx = row0 + ln;
  if (arow_idx >= B) arow_idx = B - 1;        // clamp (stores are guarded)
  const float* arow = F + (size_t)arow_idx * 784 + 2 * half;
  const float* brow = ldsW + ln * W_LDS_STRIDE + 2 * half;

  for (int k0 = 0; k0 < 784; k0 += 4) {
    v2f av, bv;
    av.x = arow[k0];
    av.y = arow[k0 + 1];
    bv.x = brow[k0];
    bv.y = brow[k0 + 1];
    c = __builtin_amdgcn_wmma_f32_16x16x4_f32(false, av, false, bv,
                                              (short)0, c, false, false);
  }
#pragma unroll
  for (int r = 0; r < 8; ++r) {
    int m = row0 + half * 8 + r;
    if (m < B) L[(size_t)m * 16 + ln] = c[r];
  }
}

// ---------------------------------------------------------------------------
// Kernel 4: row-wise log_softmax over the 10 valid logits.
// ---------------------------------------------------------------------------
__global__ void logsoftmax_kernel(const float* __restrict__ L,
                                  float* __restrict__ out, int B) {
  int row = blockIdx.x * blockDim.x + threadIdx.x;
  if (row >= B) return;
  float v[10];
  float m = -1e30f;
#pragma unroll
  for (int j = 0; j < 10; ++j) { v[j] = L[(size_t)row * 16 + j]; m = fmaxf(m, v[j]); }
  float ssum = 0.f;
#pragma unroll
  for (int j = 0; j < 10; ++j) ssum += expf(v[j] - m);
  float lse = m + logf(ssum);
#pragma unroll
  for (int j = 0; j < 10; ++j) out[(size_t)row * 10 + j] = v[j] - lse;
}

// ---------------------------------------------------------------------------
extern "C" void kernel_launch(void* const* d_in, const int* in_sizes, int n_in,
                              void* d_out, int out_size, void* d_ws, size_t ws_size,
                              hipStream_t stream) {
  const float* x    = (const float*)d_in[0];   // (B,1,28,28)
  const float* si   = (const float*)d_in[1];   // (B,2)
  const float* sw   = (const float*)d_in[2];   // (2,)
  const float* qp   = (const float*)d_in[3];   // (6,)
  const float* W    = (const float*)d_in[4];   // (10,784)
  const float* bias = (const float*)d_in[5];   // (10,)
  float* out = (float*)d_out;

  int B = in_sizes[0] / 784;

  float* ws  = (float*)d_ws;
  float* a01 = ws;                          // B*2
  float* F   = a01 + (size_t)2 * B;         // B*784 (L2-resident, 12.8 MB @ B=4096)
  float* L   = F + (size_t)B * 784;         // B*16

  sampler_kernel<<<(B + 255) / 256, 256, 0, stream>>>(si, sw, a01, B);

  int total = B * 196;
  patch_kernel<<<(total + 255) / 256, 256, 0, stream>>>(x, a01, qp, F, B);

  int waves  = (B + 15) / 16;               // one wave per 16 rows
  int blocks = (waves + 7) / 8;             // 8 waves (256 threads) per block
  gemm_wmma_kernel<<<blocks, 256, 0, stream>>>(F, W, bias, L, B);

  logsoftmax_kernel<<<(B + 255) / 256, 256, 0, stream>>>(L, out, B);
}